// ManifoldGaussian3D_5334349381677
// MI455X (gfx1250) — compile-verified
//
#include <hip/hip_runtime.h>

// ManifoldGaussian3D: cov = (R(q_norm) * diag(exp(ls))) @ (.)^T per row.
// Bandwidth-bound: 64 B/item * 4M items = 256 MB traffic, ~11 us floor @ 23.3 TB/s.
// CDNA5 path: b128 nontemporal quat loads, async global<->LDS staging (ASYNCcnt)
// for the stride-3 scales (copy-in) and the stride-9 covariances (copy-out).

#define BLOCK 256

typedef float v4f __attribute__((ext_vector_type(4)));

__device__ __forceinline__ unsigned lds_off(const void* p) {
  // low 32 bits of a generic LDS pointer == byte offset in the workgroup's LDS alloc
  return (unsigned)(unsigned long long)p;
}

__device__ __forceinline__ void gauss_cov(float w, float x, float y, float z,
                                          float l0, float l1, float l2,
                                          float* __restrict__ c) {
  float nn  = sqrtf(w * w + x * x + y * y + z * z) + 1e-8f;
  float inv = 1.0f / nn;
  w *= inv; x *= inv; y *= inv; z *= inv;

  float r00 = 1.0f - 2.0f * (y * y + z * z);
  float r01 = 2.0f * (x * y - w * z);
  float r02 = 2.0f * (x * z + w * y);
  float r10 = 2.0f * (x * y + w * z);
  float r11 = 1.0f - 2.0f * (x * x + z * z);
  float r12 = 2.0f * (y * z - w * x);
  float r20 = 2.0f * (x * z - w * y);
  float r21 = 2.0f * (y * z + w * x);
  float r22 = 1.0f - 2.0f * (x * x + y * y);

  float e0 = expf(l0), e1 = expf(l1), e2 = expf(l2);
  float sa = e0 * e0, sb = e1 * e1, sc = e2 * e2;

  float c00 = r00 * r00 * sa + r01 * r01 * sb + r02 * r02 * sc;
  float c01 = r00 * r10 * sa + r01 * r11 * sb + r02 * r12 * sc;
  float c02 = r00 * r20 * sa + r01 * r21 * sb + r02 * r22 * sc;
  float c11 = r10 * r10 * sa + r11 * r11 * sb + r12 * r12 * sc;
  float c12 = r10 * r20 * sa + r11 * r21 * sb + r12 * r22 * sc;
  float c22 = r20 * r20 * sa + r21 * r21 * sb + r22 * r22 * sc;

  c[0] = c00; c[1] = c01; c[2] = c02;
  c[3] = c01; c[4] = c11; c[5] = c12;
  c[6] = c02; c[7] = c12; c[8] = c22;
}

__global__ __launch_bounds__(BLOCK) void cov3x3_kernel(
    const float* __restrict__ quat,  // [n,4]
    const float* __restrict__ logs,  // [n,3]
    float* __restrict__ out,         // [n,3,3]
    int n) {
  __shared__ __align__(16) float lds_s[BLOCK * 3];  // staged scales
  __shared__ __align__(16) float lds_o[BLOCK * 9];  // staged covariances

  const int tid  = threadIdx.x;
  const int base = blockIdx.x * BLOCK;

  if (base + BLOCK <= n) {
    // --- async global->LDS b128 copy-in of this block's 768 scale floats
    //     (768 floats = 192 lanes * 16 B; fixes the stride-3 AoS pattern)
    const float* sg = logs + (size_t)base * 3;
    if (tid < 192) {
      asm volatile("global_load_async_to_lds_b128 %0, %1, %2 th:TH_LOAD_NT"
                   :
                   : "v"(lds_off(&lds_s[tid * 4])), "v"(tid * 16), "s"(sg)
                   : "memory");
    }

    // --- overlap: coalesced 16 B nontemporal quaternion load (streamed once)
    const v4f qv =
        __builtin_nontemporal_load(&((const v4f*)quat)[base + tid]);

    asm volatile("s_wait_asynccnt 0" ::: "memory");
    __syncthreads();  // all waves' async copies landed in LDS

    const float l0 = lds_s[tid * 3 + 0];
    const float l1 = lds_s[tid * 3 + 1];
    const float l2 = lds_s[tid * 3 + 2];

    float c[9];
    gauss_cov(qv.x, qv.y, qv.z, qv.w, l0, l1, l2, c);

#pragma unroll
    for (int j = 0; j < 9; ++j) lds_o[tid * 9 + j] = c[j];
    __syncthreads();  // LDS results visible before async engine reads them

    // --- async LDS->global b128 copy-out: 2304 floats = 576 x 16 B chunks
    const float* ob = out + (size_t)base * 9;
#pragma unroll
    for (int k = 0; k < 2; ++k) {
      const int idx = tid + k * BLOCK;
      asm volatile("global_store_async_from_lds_b128 %0, %1, %2 th:TH_STORE_NT"
                   :
                   : "v"(idx * 16), "v"(lds_off(&lds_o[idx * 4])), "s"(ob)
                   : "memory");
    }
    if (tid < 64) {
      const int idx = tid + 2 * BLOCK;
      asm volatile("global_store_async_from_lds_b128 %0, %1, %2 th:TH_STORE_NT"
                   :
                   : "v"(idx * 16), "v"(lds_off(&lds_o[idx * 4])), "s"(ob)
                   : "memory");
    }
    // s_endpgm performs an implicit wait-idle, draining ASYNCcnt.
  } else {
    // generic scalar tail (never taken for N = 4M, kept for correctness)
    const int i = base + tid;
    if (i >= n) return;
    float c[9];
    gauss_cov(quat[i * 4 + 0], quat[i * 4 + 1], quat[i * 4 + 2], quat[i * 4 + 3],
              logs[i * 3 + 0], logs[i * 3 + 1], logs[i * 3 + 2], c);
#pragma unroll
    for (int j = 0; j < 9; ++j) out[(size_t)i * 9 + j] = c[j];
  }
}

extern "C" void kernel_launch(void* const* d_in, const int* in_sizes, int n_in,
                              void* d_out, int out_size, void* d_ws, size_t ws_size,
                              hipStream_t stream) {
  const float* quat = (const float*)d_in[0];  // [N,4] float32
  const float* logs = (const float*)d_in[1];  // [N,3] float32
  float* out        = (float*)d_out;          // [N,3,3] float32

  const int n    = in_sizes[0] / 4;
  const int grid = (n + BLOCK - 1) / BLOCK;
  cov3x3_kernel<<<grid, BLOCK, 0, stream>>>(quat, logs, out, n);
}